// VectorQuantizer_83262236000460
// MI455X (gfx1250) — compile-verified
//
#include <hip/hip_runtime.h>

typedef __attribute__((ext_vector_type(2))) float v2f;
typedef __attribute__((ext_vector_type(8))) float v8f;

#define N_ROWS 32768
#define D_DIM  128
#define K_CODES 1024

// ---------------------------------------------------------------------------
// Prep 1: transpose emb [D,K] -> embT [K,D] (makes gather & scales contiguous)
// ---------------------------------------------------------------------------
__global__ __launch_bounds__(256)
void vq_transpose_kernel(const float* __restrict__ emb, float* __restrict__ embT) {
    int e = blockIdx.x * blockDim.x + threadIdx.x;   // 0 .. 131071
    int d = e >> 10;        // row of emb
    int k = e & (K_CODES - 1);
    embT[(size_t)k * D_DIM + d] = emb[e];
}

// ---------------------------------------------------------------------------
// Prep 2: ||e_k||^2 per code, and zero the (poisoned) loss accumulator slot
// ---------------------------------------------------------------------------
__global__ __launch_bounds__(256)
void vq_enorm_kernel(const float* __restrict__ embT, float* __restrict__ enorm,
                     float* __restrict__ lossAcc) {
    int k = blockIdx.x * blockDim.x + threadIdx.x;   // 0..1023
    if (k == 0) *lossAcc = 0.0f;
    const float* row = embT + (size_t)k * D_DIM;
    float s = 0.0f;
    for (int d = 0; d < D_DIM; ++d) { float v = row[d]; s += v * v; }
    enorm[k] = s;
}

// ---------------------------------------------------------------------------
// Main: fused S = x@E (fp32 WMMA) + argmin_k(||e_k||^2 - 2 S[n,k]).
// One wave32 handles a 32-row block of x (two 16x128 A fragment sets kept in
// VGPRs) so each B fragment pair feeds TWO independent WMMA chains:
//   - halves B traffic from L2 (0.5 GB total instead of 1 GB)
//   - 2 wmma per s_wait_loadcnt -> better load-latency hiding
// A fragment (16x4 f32): lanes 0-15 = rows M, v0 holds K=k0 (lo half)/k0+2 (hi),
// v1 holds K=k0+1 / k0+3  -> per lane a consecutive float pair of its x row.
// C tile (16x16 f32): lane = col N (mod 16), vgpr r = row M=r (lo) / r+8 (hi).
// ---------------------------------------------------------------------------
__global__ __launch_bounds__(128)
void vq_argmin_wmma_kernel(const float* __restrict__ x,
                           const float* __restrict__ emb,
                           const float* __restrict__ enorm,
                           int* __restrict__ indices) {
    const int lane  = threadIdx.x & 31;
    const int wave  = threadIdx.x >> 5;
    const int m     = lane & 15;     // row (A) / col (B,C) within tile
    const int half  = lane >> 4;     // 0 or 1
    const int rowBase = (blockIdx.x * 4 + wave) * 32;

    // Preload A fragments for two 16x128 x-blocks (reused for all 64 k tiles)
    v2f a0[32], a1[32];
    const float* xr0 = x + (size_t)(rowBase + m) * D_DIM;
    const float* xr1 = xr0 + 16 * D_DIM;
    #pragma unroll
    for (int s = 0; s < 32; ++s) {
        const int kk = 4 * s + 2 * half;
        v2f t0; t0.x = xr0[kk]; t0.y = xr0[kk + 1]; a0[s] = t0;
        v2f t1; t1.x = xr1[kk]; t1.y = xr1[kk + 1]; a1[s] = t1;
    }

    float bv0[8], bv1[8]; int bi0[8], bi1[8];
    #pragma unroll
    for (int r = 0; r < 8; ++r) {
        bv0[r] = 3.0e38f; bi0[r] = 0;
        bv1[r] = 3.0e38f; bi1[r] = 0;
    }

    for (int kt = 0; kt < K_CODES; kt += 16) {
        const int col = kt + m;                 // code index this lane owns
        v8f acc0 = {0.f, 0.f, 0.f, 0.f, 0.f, 0.f, 0.f, 0.f};
        v8f acc1 = {0.f, 0.f, 0.f, 0.f, 0.f, 0.f, 0.f, 0.f};
        #pragma unroll
        for (int s = 0; s < 32; ++s) {
            const int kd = 4 * s + 2 * half;    // mirror A's K striping for B
            v2f b;
            b.x = emb[(size_t)kd * K_CODES + col];
            b.y = emb[(size_t)(kd + 1) * K_CODES + col];
            acc0 = __builtin_amdgcn_wmma_f32_16x16x4_f32(
                       false, a0[s], false, b, (short)0, acc0, false, false);
            acc1 = __builtin_amdgcn_wmma_f32_16x16x4_f32(
                       false, a1[s], false, b, (short)0, acc1, false, false);
        }
        const float en = enorm[col];
        #pragma unroll
        for (int r = 0; r < 8; ++r) {
            const float d0 = en - 2.0f * acc0[r];   // ||x||^2 dropped: const/row
            const float d1 = en - 2.0f * acc1[r];
            if (d0 < bv0[r]) { bv0[r] = d0; bi0[r] = col; }  // strict <: first-min
            if (d1 < bv1[r]) { bv1[r] = d1; bi1[r] = col; }
        }
    }

    // Cross-lane argmin within each 16-lane half (half 0: rows r, half 1: r+8)
    #pragma unroll
    for (int r = 0; r < 8; ++r) {
        #pragma unroll
        for (int off = 8; off >= 1; off >>= 1) {
            float ov; int oi;
            ov = __shfl_xor(bv0[r], off, 16);
            oi = __shfl_xor(bi0[r], off, 16);
            if (ov < bv0[r] || (ov == bv0[r] && oi < bi0[r])) { bv0[r] = ov; bi0[r] = oi; }
            ov = __shfl_xor(bv1[r], off, 16);
            oi = __shfl_xor(bi1[r], off, 16);
            if (ov < bv1[r] || (ov == bv1[r] && oi < bi1[r])) { bv1[r] = ov; bi1[r] = oi; }
        }
    }
    if (m == 0) {
        #pragma unroll
        for (int r = 0; r < 8; ++r) {
            indices[rowBase + r + 8 * half]      = bi0[r];
            indices[rowBase + 16 + r + 8 * half] = bi1[r];
        }
    }
}

// ---------------------------------------------------------------------------
// Gather quantized rows (contiguous via embT) + loss = 1.25 * mean((q-x)^2)
// (commitment 0.25*mean + codebook 1.0*mean share the same forward value;
//  straight-through output forward value == quantized). float4 path: each
//  thread owns 4 consecutive d of a single row -> b128 loads/stores.
// ---------------------------------------------------------------------------
__global__ __launch_bounds__(256)
void vq_gather_loss_kernel(const float* __restrict__ x,
                           const float* __restrict__ embT,
                           const int* __restrict__ indices,
                           float* __restrict__ outQ,
                           float* __restrict__ lossAcc) {
    __shared__ float red[256];
    const int tid = threadIdx.x;
    const size_t base = (size_t)blockIdx.x * 1024 + (size_t)tid * 4;
    const int n = (int)(base >> 7);       // / 128 (4 consecutive d, same row)
    const int d = (int)(base & 127);

    const float4 q  = *(const float4*)(embT + (size_t)indices[n] * D_DIM + d);
    const float4 xv = *(const float4*)(x + base);
    *(float4*)(outQ + base) = q;

    const float dx = q.x - xv.x, dy = q.y - xv.y, dz = q.z - xv.z, dw = q.w - xv.w;
    red[tid] = dx * dx + dy * dy + dz * dz + dw * dw;
    __syncthreads();
    for (int off = 128; off > 0; off >>= 1) {
        if (tid < off) red[tid] += red[tid + off];
        __syncthreads();
    }
    if (tid == 0)
        atomicAdd(lossAcc, red[0] * (1.25f / (float)(N_ROWS * D_DIM)));
}

// ---------------------------------------------------------------------------
// Per-code nearest-other-code distance (second order statistic of the full
// row, incl. the ~0 self-distance, matching sort(...)[:,1]) -> scales / (D/2)
// ---------------------------------------------------------------------------
__global__ __launch_bounds__(128)
void vq_scales_kernel(const float* __restrict__ embT,
                      const float* __restrict__ enorm,
                      float* __restrict__ scales) {
    __shared__ float ci[D_DIM];
    __shared__ float s1a[128], s2a[128];
    const int i = blockIdx.x;
    const int tid = threadIdx.x;
    ci[tid] = embT[(size_t)i * D_DIM + tid];
    __syncthreads();

    const float eni = enorm[i];
    float s1 = 3.0e38f, s2 = 3.0e38f;     // two smallest, s1 <= s2
    for (int j = tid; j < K_CODES; j += 128) {
        const float* cj = embT + (size_t)j * D_DIM;
        float dot = 0.0f;
        #pragma unroll 8
        for (int d = 0; d < D_DIM; ++d) dot += ci[d] * cj[d];
        const float dist = eni + enorm[j] - 2.0f * dot;
        if (dist < s1) { s2 = s1; s1 = dist; }
        else if (dist < s2) { s2 = dist; }
    }
    s1a[tid] = s1; s2a[tid] = s2;
    __syncthreads();
    for (int off = 64; off > 0; off >>= 1) {
        if (tid < off) {
            const float a1 = s1a[tid], a2 = s2a[tid];
            const float b1 = s1a[tid + off], b2 = s2a[tid + off];
            s1a[tid] = fminf(a1, b1);
            s2a[tid] = fminf(fmaxf(a1, b1), fminf(a2, b2));
        }
        __syncthreads();
    }
    const float s = s2a[0] * (1.0f / (D_DIM / 2));
    scales[(size_t)i * D_DIM + tid] = s;
}

// ---------------------------------------------------------------------------
extern "C" void kernel_launch(void* const* d_in, const int* in_sizes, int n_in,
                              void* d_out, int out_size, void* d_ws, size_t ws_size,
                              hipStream_t stream) {
    (void)in_sizes; (void)n_in; (void)out_size; (void)ws_size;
    const float* x   = (const float*)d_in[0];   // [64,512,128]
    const float* emb = (const float*)d_in[1];   // [128,1024]

    float* outQ   = (float*)d_out;                          // 4,194,304 floats
    float* loss   = (float*)d_out + (size_t)N_ROWS * D_DIM; // 1 float
    float* scales = loss + 1;                               // 131,072 floats

    // workspace: enorm[1024] | indices[32768] | embT[131072]
    float* enorm   = (float*)d_ws;
    int*   indices = (int*)((float*)d_ws + K_CODES);
    float* embT    = (float*)d_ws + K_CODES + N_ROWS;

    vq_transpose_kernel  <<<512, 256, 0, stream>>>(emb, embT);
    vq_enorm_kernel      <<<4, 256, 0, stream>>>(embT, enorm, loss);
    vq_argmin_wmma_kernel<<<256, 128, 0, stream>>>(x, emb, enorm, indices);
    vq_gather_loss_kernel<<<4096, 256, 0, stream>>>(x, embT, indices, outQ, loss);
    vq_scales_kernel     <<<K_CODES, 128, 0, stream>>>(embT, enorm, scales);
}